// AnglePotentials_40535901339791
// MI455X (gfx1250) — compile-verified
//
#include <hip/hip_runtime.h>
#include <hip/hip_bf16.h>

typedef __attribute__((ext_vector_type(2))) float v2f;
typedef __attribute__((ext_vector_type(8))) float v8f;

// ---------------------------------------------------------------------------
// Wave32 sum-reduction using the CDNA5 matrix pipe.
// A = per-lane partial sums placed in A-VGPR0 (doc'd 16x4 f32 A layout:
//   lanes 0-15 hold A[m][0..1], lanes 16-31 hold A[m][2..3], m = lane%16),
// B = all-ones (layout-independent), C = 0.
//   => D[m][n] = s_m + s_{m+16}  (uniform across n)
// D layout: VGPR j holds row j (lanes 0-15) and row j+8 (lanes 16-31).
// Sum the 8 D VGPRs per lane, then fold the two lane-halves with shfl_xor(16).
// Every lane ends with the exact f32 wave total. EXEC must be all ones here.
// ---------------------------------------------------------------------------
__device__ __forceinline__ float wave_reduce32(float s) {
#if __has_builtin(__builtin_amdgcn_wmma_f32_16x16x4_f32)
    v2f a; a[0] = s;    a[1] = 0.0f;
    v2f b; b[0] = 1.0f; b[1] = 1.0f;
    v8f c = {0.f, 0.f, 0.f, 0.f, 0.f, 0.f, 0.f, 0.f};
    v8f d = __builtin_amdgcn_wmma_f32_16x16x4_f32(
        /*neg_a=*/false, a, /*neg_b=*/false, b,
        /*c_mod=*/(short)0, c, /*reuse_a=*/false, /*reuse_b=*/false);
    float t = ((d[0] + d[1]) + (d[2] + d[3])) + ((d[4] + d[5]) + (d[6] + d[7]));
    t += __shfl_xor(t, 16, 32);
    return t;
#else
    for (int off = 16; off > 0; off >>= 1) s += __shfl_xor(s, off, 32);
    return s;
#endif
}

// minimum-image wrap matching: v + (-(v>=0.5c) + (v<-0.5c)) * c
__device__ __forceinline__ float pbc_wrap(float v, float c, float h) {
    float o = (v < -h ? 1.0f : 0.0f) - (v >= h ? 1.0f : 0.0f);
    return fmaf(o, c, v);
}

__global__ __launch_bounds__(256) void angle_energy_partial(
    const float* __restrict__ xyz,
    const int*   __restrict__ top,
    const float* __restrict__ cell,
    const float* __restrict__ thetao_p,
    int n_angles,
    float* __restrict__ partial)
{
    const float cx = cell[0], cy = cell[1], cz = cell[2];
    const float hx = 0.5f * cx, hy = 0.5f * cy, hz = 0.5f * cz;
    const float th0 = thetao_p[0];

    float acc = 0.0f;
    const int stride = gridDim.x * blockDim.x;
    for (int idx = blockIdx.x * blockDim.x + threadIdx.x; idx < n_angles; idx += stride) {
        const size_t base = 3u * (size_t)idx;
        // index stream is read exactly once: non-temporal so xyz stays hot in L2
        const int i0 = __builtin_nontemporal_load(top + base + 0);
        const int i1 = __builtin_nontemporal_load(top + base + 1);
        const int i2 = __builtin_nontemporal_load(top + base + 2);

        const size_t b0 = 3u * (size_t)(unsigned)i0;
        const size_t b1 = 3u * (size_t)(unsigned)i1;
        const size_t b2 = 3u * (size_t)(unsigned)i2;

        const float p0x = xyz[b0], p0y = xyz[b0 + 1], p0z = xyz[b0 + 2];
        const float p1x = xyz[b1], p1y = xyz[b1 + 1], p1z = xyz[b1 + 2];
        const float p2x = xyz[b2], p2y = xyz[b2 + 1], p2z = xyz[b2 + 2];

        const float ax = pbc_wrap(p0x - p1x, cx, hx);
        const float ay = pbc_wrap(p0y - p1y, cy, hy);
        const float az = pbc_wrap(p0z - p1z, cz, hz);
        const float bx = pbc_wrap(p2x - p1x, cx, hx);
        const float by = pbc_wrap(p2y - p1y, cy, hy);
        const float bz = pbc_wrap(p2z - p1z, cz, hz);

        const float dot = ax * bx + ay * by + az * bz;
        const float na  = ax * ax + ay * ay + az * az;
        const float nb  = bx * bx + by * by + bz * bz;

        float t = dot / sqrtf(na * nb);
        t = fminf(1.0f, fmaxf(-1.0f, t));   // guard rounding past +/-1
        const float dth = acosf(t) - th0;
        acc += dth * dth;
    }

    // all threads reconverged here -> EXEC all ones for WMMA
    const float wsum = wave_reduce32(acc);

    __shared__ float smem[8];
    const int tid = threadIdx.x;
    if ((tid & 31) == 0) smem[tid >> 5] = wsum;
    __syncthreads();
    if (tid == 0) {
        float r = 0.0f;
        const int nw = blockDim.x >> 5;
        for (int i = 0; i < nw; ++i) r += smem[i];
        partial[blockIdx.x] = r;
    }
}

__global__ __launch_bounds__(256) void angle_energy_final(
    const float* __restrict__ partial,
    int nb,
    const float* __restrict__ k_p,
    float* __restrict__ out)
{
    float acc = 0.0f;
    for (int i = threadIdx.x; i < nb; i += blockDim.x) acc += partial[i];

    const float wsum = wave_reduce32(acc);

    __shared__ float smem[8];
    const int tid = threadIdx.x;
    if ((tid & 31) == 0) smem[tid >> 5] = wsum;
    __syncthreads();
    if (tid == 0) {
        float r = 0.0f;
        const int nw = blockDim.x >> 5;
        for (int i = 0; i < nw; ++i) r += smem[i];
        out[0] = 0.5f * k_p[0] * r;
    }
}

extern "C" void kernel_launch(void* const* d_in, const int* in_sizes, int n_in,
                              void* d_out, int out_size, void* d_ws, size_t ws_size,
                              hipStream_t stream) {
    const float* xyz    = (const float*)d_in[0];
    const int*   top    = (const int*)  d_in[1];
    const float* cell   = (const float*)d_in[2];
    const float* k_p    = (const float*)d_in[3];
    const float* th_p   = (const float*)d_in[4];

    const int n_angles = in_sizes[1] / 3;

    float* partial = (float*)d_ws;
    int nb = 2048;
    const int maxnb = (int)(ws_size / sizeof(float));
    if (maxnb > 0 && nb > maxnb) nb = maxnb;
    if (nb < 1) nb = 1;

    angle_energy_partial<<<nb, 256, 0, stream>>>(xyz, top, cell, th_p, n_angles, partial);
    angle_energy_final<<<1, 256, 0, stream>>>(partial, nb, k_p, (float*)d_out);
}